// BasicLP_75453985456747
// MI455X (gfx1250) — compile-verified
//
#include <hip/hip_runtime.h>
#include <math.h>

// Batched 32x32 LP interior-point solver for MI455X (gfx1250, wave32).
// One wave per LP instance. A kept in bank-conflict-free padded LDS,
// state vectors one element per lane. The normal-matrix formation
// M = A^T diag(w1) A uses V_WMMA_F32_16X16X4_F32 (fp32 matrix cores).

typedef float v2f __attribute__((ext_vector_type(2)));
typedef float v8f __attribute__((ext_vector_type(8)));

#define N_ITER 40
#define SIGMA  0.1f
#define REGEPS 1e-8f
#define WPB    4          // waves (LP instances) per block
#define LDSROW 33         // padded row stride: conflict-free rows & columns

__device__ __forceinline__ void wave_sync() {
  // Wave-private LDS RAW ordering: DS ops are in-order per wave; just make
  // sure issued DS writes complete and the compiler doesn't reorder.
  __builtin_amdgcn_wave_barrier();
  __asm__ volatile("s_wait_dscnt 0" ::: "memory");
  __builtin_amdgcn_wave_barrier();
}

__device__ __forceinline__ float wave_sum(float v) {
  #pragma unroll
  for (int o = 16; o > 0; o >>= 1) v += __shfl_xor(v, o, 32);
  return v;
}
__device__ __forceinline__ float wave_min(float v) {
  #pragma unroll
  for (int o = 16; o > 0; o >>= 1) v = fminf(v, __shfl_xor(v, o, 32));
  return v;
}
__device__ __forceinline__ float ratio_term(float v, float dv) {
  return (dv < 0.0f) ? (-v / dv) : INFINITY;
}

// y[t] = sum_j A[t][j] * x[j]   (lane t holds row t result)
__device__ __forceinline__ float matvec_row(const float* lA, int t, float xv) {
  float acc = 0.0f;
  #pragma unroll 8
  for (int j = 0; j < 32; ++j)
    acc = fmaf(lA[t * LDSROW + j], __shfl(xv, j, 32), acc);
  return acc;
}
// y[t] = sum_k A[k][t] * z[k]   (A^T matvec)
__device__ __forceinline__ float matvec_col(const float* lA, int t, float zv) {
  float acc = 0.0f;
  #pragma unroll 8
  for (int k = 0; k < 32; ++k)
    acc = fmaf(lA[k * LDSROW + t], __shfl(zv, k, 32), acc);
  return acc;
}

__global__ void lp_zero_out(float* o) { o[0] = 0.0f; }

__global__ __launch_bounds__(WPB * 32)
void lp_ipm_kernel(const float* __restrict__ Ag,
                   const float* __restrict__ bg,
                   const float* __restrict__ cg,
                   float* __restrict__ out, int B) {
  __shared__ float sA[WPB][32 * LDSROW];
  __shared__ float sM[WPB][32 * LDSROW];

  const int lane  = threadIdx.x & 31;
  const int w     = threadIdx.x >> 5;
  const int batch = blockIdx.x * WPB + w;
  if (batch >= B) return;           // wave-uniform exit (EXEC stays all-1s)

  float* lA = sA[w];
  float* lM = sM[w];
  const int t    = lane;
  const int half = lane >> 4;       // 0 or 1 (which 16-lane half)
  const int li   = lane & 15;

  // Load A (row-major 32x32) into padded LDS; coalesced per row.
  const float* Ab = Ag + (size_t)batch * 32 * 32;
  #pragma unroll 8
  for (int r = 0; r < 32; ++r)
    lA[r * LDSROW + lane] = Ab[r * 32 + lane];
  wave_sync();

  const float bv = bg[(size_t)batch * 32 + lane];
  const float cv = cg[(size_t)batch * 32 + lane];
  const float q  = -cv;

  float x = 1.0f, s1 = 1.0f, z1 = 1.0f, s2 = 1.0f, z2 = 1.0f;

  #pragma unroll 1
  for (int it = 0; it < N_ITER; ++it) {
    const float w1 = z1 / s1;
    const float w2 = z2 / s2;

    const float Axv = matvec_row(lA, t, x);
    const float Atz = matvec_col(lA, t, z1);
    const float r_d  = q + Atz - z2;
    const float r_p1 = Axv + s1 - bv;
    const float r_p2 = -x + s2;

    const float mu  = wave_sum(s1 * z1 + s2 * z2) * (1.0f / 64.0f);
    const float rc1 = SIGMA * mu - s1 * z1;
    const float rc2 = SIGMA * mu - s2 * z2;
    const float g1  = rc1 / s1 + w1 * r_p1;
    const float g2  = rc2 / s2 + w2 * r_p2;

    const float Atg1 = matvec_col(lA, t, g1);
    const float rhs  = -r_d - Atg1 + g2;

    // ---- M = A^T diag(w1) A via fp32 WMMA 16x16x4; four 16x16 tiles, K=32.
    // A-operand (16x4 f32) layout: lanes 0-15 hold tile-row li with K={k0,k0+1}
    // in the two VGPRs; lanes 16-31 hold K={k0+2,k0+3}. B-operand mirrors it.
    v8f D00 = {0,0,0,0,0,0,0,0};
    v8f D01 = D00, D10 = D00, D11 = D00;
    #pragma unroll
    for (int k0 = 0; k0 < 32; k0 += 4) {
      v2f a0, a1, b0, b1;
      #pragma unroll
      for (int v = 0; v < 2; ++v) {
        const int   k  = k0 + half * 2 + v;
        const float wk = __shfl(w1, k, 32);
        const float e0 = lA[k * LDSROW + li];        // A[k][0..15]
        const float e1 = lA[k * LDSROW + 16 + li];   // A[k][16..31]
        a0[v] = e0;        // rows of A^T, tile I=0
        a1[v] = e1;        // rows of A^T, tile I=1
        b0[v] = wk * e0;   // diag(w1)*A, tile J=0
        b1[v] = wk * e1;   // diag(w1)*A, tile J=1
      }
      D00 = __builtin_amdgcn_wmma_f32_16x16x4_f32(false, a0, false, b0,
                                                  (short)0, D00, false, false);
      D01 = __builtin_amdgcn_wmma_f32_16x16x4_f32(false, a0, false, b1,
                                                  (short)0, D01, false, false);
      D10 = __builtin_amdgcn_wmma_f32_16x16x4_f32(false, a1, false, b0,
                                                  (short)0, D10, false, false);
      D11 = __builtin_amdgcn_wmma_f32_16x16x4_f32(false, a1, false, b1,
                                                  (short)0, D11, false, false);
    }
    // C/D layout: VGPR p -> row p (lanes 0-15) / row p+8 (lanes 16-31), N=li.
    #pragma unroll
    for (int p = 0; p < 8; ++p) {
      const int r = p + half * 8;
      lM[(r     ) * LDSROW + li     ] = D00[p];
      lM[(r     ) * LDSROW + li + 16] = D01[p];
      lM[(r + 16) * LDSROW + li     ] = D10[p];
      lM[(r + 16) * LDSROW + li + 16] = D11[p];
    }
    wave_sync();
    // Regularized diagonal and RHS column (col 32 of padded rows).
    lM[t * LDSROW + t]  += w2 + REGEPS;
    lM[t * LDSROW + 32]  = rhs;
    wave_sync();

    // ---- Solve M dx = rhs: per-wave Gaussian elimination (M is SPD+reg).
    // Lane t owns row t; pivot row k is broadcast-read from LDS.
    for (int k = 0; k < 32; ++k) {
      const float piv = lM[k * LDSROW + k];
      const float f   = (t > k) ? (lM[t * LDSROW + k] / piv) : 0.0f;
      wave_sync();
      for (int j = k; j <= 32; ++j)
        lM[t * LDSROW + j] -= f * lM[k * LDSROW + j];
      wave_sync();
    }
    // Back substitution on the rhs column.
    for (int k = 31; k >= 0; --k) {
      if (t == k) lM[k * LDSROW + 32] /= lM[k * LDSROW + k];
      wave_sync();
      const float xk = lM[k * LDSROW + 32];
      if (t < k) lM[t * LDSROW + 32] -= lM[t * LDSROW + k] * xk;
      wave_sync();
    }
    const float dx = lM[t * LDSROW + 32];

    const float Adx = matvec_row(lA, t, dx);
    const float ds1 = -r_p1 - Adx;
    const float ds2 = -r_p2 + dx;
    const float dz1 = g1 + w1 * Adx;
    const float dz2 = g2 - w2 * dx;

    float r = fminf(fminf(ratio_term(s1, ds1), ratio_term(z1, dz1)),
                    fminf(ratio_term(s2, ds2), ratio_term(z2, dz2)));
    const float alpha = 0.99f * fminf(1.0f, wave_min(r));

    x  += alpha * dx;
    s1 += alpha * ds1;
    z1 += alpha * dz1;
    s2 += alpha * ds2;
    z2 += alpha * dz2;
  }

  const float partial = wave_sum(cv * x);
  if (lane == 0) atomicAdd(out, partial);
}

extern "C" void kernel_launch(void* const* d_in, const int* in_sizes, int n_in,
                              void* d_out, int out_size, void* d_ws, size_t ws_size,
                              hipStream_t stream) {
  const float* A = (const float*)d_in[0];   // [B,32,32]
  const float* b = (const float*)d_in[1];   // [B,32]
  const float* c = (const float*)d_in[2];   // [B,32]
  float* out = (float*)d_out;               // scalar

  const int B = in_sizes[1] / 32;           // 8192

  lp_zero_out<<<1, 1, 0, stream>>>(out);
  const int blocks = (B + WPB - 1) / WPB;
  lp_ipm_kernel<<<blocks, WPB * 32, 0, stream>>>(A, b, c, out, B);
}